// CustomGATLayerEdge_51788715655857
// MI455X (gfx1250) — compile-verified
//
#include <hip/hip_runtime.h>
#include <math.h>

typedef __attribute__((ext_vector_type(2))) float v2f;
typedef __attribute__((ext_vector_type(8))) float v8f;

#define D_IN  64
#define D_OUT 16
#define NH    4      // heads
#define HD    64     // NH * D_OUT
#define EPS   1e-5f

// ---------------- init ----------------
__global__ void k_zero(float* __restrict__ p, int n) {
  int i = blockIdx.x * blockDim.x + threadIdx.x;
  if (i < n) p[i] = 0.f;
}

// ---------------- fold W_e into q[h][64] = W_e[h] @ W_attn[h,0:16] ----------------
__global__ void k_q(const float* __restrict__ We, const float* __restrict__ Wattn,
                    float* __restrict__ q) {
  int i = threadIdx.x;          // 256 threads: h = i/64, k = i%64
  int h = i >> 6, k = i & 63;
  float acc = 0.f;
  #pragma unroll
  for (int o = 0; o < D_OUT; ++o)
    acc += We[h * (D_IN * D_OUT) + k * D_OUT + o] * Wattn[h * 48 + o];
  q[h * 64 + k] = acc;
}

// ---------------- z_h = h @ W_h per head, fp32 WMMA 16x16x4 ----------------
// one wave (32 threads) per (16-node tile, head). K=64 chained in 16 WMMA steps.
// A 16x4 f32: lane 0-15 -> M=lane, K={0,1}; lane 16-31 -> M=lane-16, K={2,3}
// B 4x16 f32: lane -> N=lane%16, K = 2*(lane/16) + vgpr
// C/D 16x16:  lane -> N=lane%16, M = vgpr + 8*(lane/16)
__global__ void k_proj(const float* __restrict__ hfeat,
                       const float* __restrict__ Wh,
                       float* __restrict__ zh) {
  const int tile = blockIdx.x;
  const int head = blockIdx.y;
  const int lane = threadIdx.x;
  const int lo = lane & 15;
  const int hi = lane >> 4;
  const float* __restrict__ Wbase = Wh + head * (D_IN * D_OUT);
  const int mrow = tile * 16 + lo;

  v8f c = {0.f, 0.f, 0.f, 0.f, 0.f, 0.f, 0.f, 0.f};
  #pragma unroll
  for (int k0 = 0; k0 < D_IN; k0 += 4) {
    v2f a = *(const v2f*)(hfeat + mrow * D_IN + k0 + 2 * hi);
    v2f b;
    b.x = Wbase[(k0 + 2 * hi) * D_OUT + lo];
    b.y = Wbase[(k0 + 2 * hi + 1) * D_OUT + lo];
    c = __builtin_amdgcn_wmma_f32_16x16x4_f32(false, a, false, b,
                                              (short)0, c, false, false);
  }
  #pragma unroll
  for (int v = 0; v < 8; ++v) {
    int m = tile * 16 + v + 8 * hi;
    zh[m * HD + head * D_OUT + lo] = c[v];
  }
}

// scalar tail for N % 16 rows (keeps WMMA kernel EXEC=all-ones)
__global__ void k_proj_tail(const float* __restrict__ hfeat,
                            const float* __restrict__ Wh,
                            float* __restrict__ zh, int n0, int Nn) {
  int i = blockIdx.x * blockDim.x + threadIdx.x;   // over (Nn-n0)*HD
  int total = (Nn - n0) * HD;
  if (i >= total) return;
  int n = n0 + i / HD, c = i % HD, h = c >> 4, o = c & 15;
  float acc = 0.f;
  for (int k = 0; k < D_IN; ++k)
    acc += hfeat[n * D_IN + k] * Wh[h * (D_IN * D_OUT) + k * D_OUT + o];
  zh[n * HD + c] = acc;
}

// ---------------- s1[n,h] = z_h . W_attn[h,16:32], s2 with 32:48 ----------------
__global__ void k_s12(const float* __restrict__ zh, const float* __restrict__ Wattn,
                      float* __restrict__ s1, float* __restrict__ s2, int Nn) {
  int i = blockIdx.x * blockDim.x + threadIdx.x;   // over N*NH
  if (i >= Nn * NH) return;
  int n = i / NH, h = i % NH;
  const float* z = zh + n * HD + h * D_OUT;
  const float* w1 = Wattn + h * 48 + 16;
  const float* w2 = Wattn + h * 48 + 32;
  float a = 0.f, b = 0.f;
  #pragma unroll
  for (int o = 0; o < D_OUT; ++o) { a += z[o] * w1[o]; b += z[o] * w2[o]; }
  s1[i] = a; s2[i] = b;
}

// ---------------- edge logits + encoded segment max ----------------
__global__ void k_attn(const float* __restrict__ efeat,
                       const int* __restrict__ src, const int* __restrict__ dst,
                       const float* __restrict__ q,
                       const float* __restrict__ s1, const float* __restrict__ s2,
                       float* __restrict__ attn, unsigned* __restrict__ mEnc, int E) {
  __shared__ float qs[NH * 64];
  if (threadIdx.x < NH * 64) qs[threadIdx.x] = q[threadIdx.x];
  __syncthreads();
  int e = blockIdx.x * blockDim.x + threadIdx.x;
  if (e >= E) return;
  const float* __restrict__ row = efeat + (long)e * D_IN;
  float d[NH] = {0.f, 0.f, 0.f, 0.f};
  for (int k = 0; k < D_IN; k += 4) {
    float4 x = *(const float4*)(row + k);
    #pragma unroll
    for (int h = 0; h < NH; ++h) {
      const float* qh = qs + h * 64 + k;
      d[h] += x.x * qh[0] + x.y * qh[1] + x.z * qh[2] + x.w * qh[3];
    }
  }
  int s = src[e], t = dst[e];
  #pragma unroll
  for (int h = 0; h < NH; ++h) {
    float a = d[h] + s1[s * NH + h] + s2[t * NH + h];
    a = (a > 0.f) ? a : 0.01f * a;                 // leaky_relu(0.01)
    attn[(long)e * NH + h] = a;
    unsigned u = __float_as_uint(a);
    unsigned key = (u & 0x80000000u) ? ~u : (u | 0x80000000u);
    atomicMax(&mEnc[t * NH + h], key);
  }
}

// ---------------- decode encoded max; empty segment -> 0 ----------------
__global__ void k_decode(unsigned* __restrict__ m, int n) {
  int i = blockIdx.x * blockDim.x + threadIdx.x;
  if (i >= n) return;
  unsigned k = m[i];
  float f = (k & 0x80000000u) ? __uint_as_float(k & 0x7FFFFFFFu)
                              : __uint_as_float(~k);
  if (!isfinite(f)) f = 0.f;
  ((float*)m)[i] = f;
}

// ---------------- a = exp(attn - m[dst]); segment sum ----------------
__global__ void k_exp(const int* __restrict__ dst, const float* __restrict__ m,
                      float* __restrict__ attn, float* __restrict__ ssum, int E) {
  int e = blockIdx.x * blockDim.x + threadIdx.x;
  if (e >= E) return;
  int t = dst[e];
  #pragma unroll
  for (int h = 0; h < NH; ++h) {
    float a = __expf(attn[(long)e * NH + h] - m[t * NH + h]);
    attn[(long)e * NH + h] = a;
    atomicAdd(&ssum[t * NH + h], a);
  }
}

// ---------------- h_agg[dst] += alpha * z_h[src] ----------------
__global__ void k_agg(const int* __restrict__ src, const int* __restrict__ dst,
                      const float* __restrict__ attn, const float* __restrict__ ssum,
                      const float* __restrict__ zh, float* __restrict__ hagg, int E) {
  int e = blockIdx.x * blockDim.x + threadIdx.x;
  if (e >= E) return;
  int s = src[e], t = dst[e];
  #pragma unroll
  for (int h = 0; h < NH; ++h) {
    float alpha = attn[(long)e * NH + h] / ssum[t * NH + h];
    const float* zs = zh + s * HD + h * D_OUT;
    float* ha = hagg + t * HD + h * D_OUT;
    #pragma unroll
    for (int o = 0; o < D_OUT; ++o) atomicAdd(&ha[o], alpha * zs[o]);
  }
}

// ---------------- per-channel (h,o) batch stats over N ----------------
__global__ void k_bnstats(const float* __restrict__ hagg,
                          float* __restrict__ mean, float* __restrict__ rstd, int Nn) {
  int c = blockIdx.x;                              // 0..63
  __shared__ float ssm[256], ssq[256];
  float s = 0.f, q = 0.f;
  for (int n = threadIdx.x; n < Nn; n += blockDim.x) {
    float v = hagg[n * HD + c];
    s += v; q += v * v;
  }
  ssm[threadIdx.x] = s; ssq[threadIdx.x] = q;
  __syncthreads();
  for (int st = 128; st > 0; st >>= 1) {
    if (threadIdx.x < st) {
      ssm[threadIdx.x] += ssm[threadIdx.x + st];
      ssq[threadIdx.x] += ssq[threadIdx.x + st];
    }
    __syncthreads();
  }
  if (threadIdx.x == 0) {
    float mu = ssm[0] / (float)Nn;
    float var = ssq[0] / (float)Nn - mu * mu;      // biased, matches jnp.var
    mean[c] = mu;
    rstd[c] = rsqrtf(var + EPS);
  }
}

// ---------------- BN affine + ELU + head-sum ----------------
__global__ void k_out(const float* __restrict__ hagg,
                      const float* __restrict__ mean, const float* __restrict__ rstd,
                      const float* __restrict__ gamma, const float* __restrict__ beta,
                      float* __restrict__ out, int Nn) {
  int i = blockIdx.x * blockDim.x + threadIdx.x;   // over N*16
  if (i >= Nn * D_OUT) return;
  int n = i >> 4, o = i & 15;
  float acc = 0.f;
  #pragma unroll
  for (int h = 0; h < NH; ++h) {
    int c = h * D_OUT + o;
    float x = (hagg[n * HD + c] - mean[c]) * rstd[c] * gamma[c] + beta[c];
    acc += (x > 0.f) ? x : (__expf(x) - 1.f);      // ELU
  }
  out[i] = acc;
}

extern "C" void kernel_launch(void* const* d_in, const int* in_sizes, int n_in,
                              void* d_out, int out_size, void* d_ws, size_t ws_size,
                              hipStream_t stream) {
  const float* h_f   = (const float*)d_in[0];
  const float* e_f   = (const float*)d_in[1];
  const int*   src   = (const int*)d_in[2];
  const int*   dst   = (const int*)d_in[3];
  const float* W_h   = (const float*)d_in[4];
  const float* W_e   = (const float*)d_in[5];
  const float* W_at  = (const float*)d_in[6];
  const float* gamma = (const float*)d_in[7];
  const float* beta  = (const float*)d_in[8];
  float* out = (float*)d_out;

  const int N = in_sizes[0] / D_IN;
  const int E = in_sizes[2];
  const int NHn = N * NH;

  // workspace layout (floats)
  float* ws   = (float*)d_ws;
  float* zh   = ws;                       // N*64
  float* q    = zh + (size_t)N * HD;      // 256
  float* s1   = q + NH * 64;              // N*NH
  float* s2   = s1 + NHn;                 // N*NH
  float* m    = s2 + NHn;                 // N*NH  (uint-encoded max, then float)
  float* ssum = m + NHn;                  // N*NH
  float* hagg = ssum + NHn;               // N*64
  float* attn = hagg + (size_t)N * HD;    // E*NH
  float* mean = attn + (size_t)E * NH;    // 64
  float* rstd = mean + HD;                // 64

  const int B = 256;
  // zero [m | ssum | hagg] contiguous region: N*NH*2 + N*64
  int nz = NHn * 2 + N * HD;
  k_zero<<<(nz + B - 1) / B, B, 0, stream>>>(m, nz);

  k_q<<<1, 256, 0, stream>>>(W_e, W_at, q);

  int tiles = N / 16;
  if (tiles > 0)
    k_proj<<<dim3(tiles, NH), 32, 0, stream>>>(h_f, W_h, zh);
  if (N % 16) {
    int n0 = tiles * 16, rem = (N - n0) * HD;
    k_proj_tail<<<(rem + B - 1) / B, B, 0, stream>>>(h_f, W_h, zh, n0, N);
  }

  k_s12<<<(NHn + B - 1) / B, B, 0, stream>>>(zh, W_at, s1, s2, N);
  k_attn<<<(E + B - 1) / B, B, 0, stream>>>(e_f, src, dst, q, s1, s2, attn,
                                            (unsigned*)m, E);
  k_decode<<<(NHn + B - 1) / B, B, 0, stream>>>((unsigned*)m, NHn);
  k_exp<<<(E + B - 1) / B, B, 0, stream>>>(dst, m, attn, ssum, E);
  k_agg<<<(E + B - 1) / B, B, 0, stream>>>(src, dst, attn, ssum, zh, hagg, E);
  k_bnstats<<<HD, 256, 0, stream>>>(hagg, mean, rstd, N);
  k_out<<<(N * D_OUT + B - 1) / B, B, 0, stream>>>(hagg, mean, rstd, gamma, beta,
                                                   out, N);
}